// FrozenBNBEmbedding_14293651161611
// MI455X (gfx1250) — compile-verified
//
#include <hip/hip_runtime.h>

// FrozenBNBEmbedding forward for MI455X (gfx1250, wave32).
// out[t, d] = code[weight_codes[input_ids[t], d] & 0xFF] * absmax[input_ids[t]]
// (BLOCKSIZE == D == 4096 => absmax block index of vocab row i is exactly i.)
//
// Pure bandwidth-bound gather/dequant: 128 MB gathered int32 codes in +
// 128 MB fp32 out => ~11 us floor at 23.3 TB/s. No matrix FLOPs exist, so no
// WMMA; the gfx1250-specific paths are global_load_async_to_lds_b32 (+
// s_wait_asynccnt) staging of the codebook, ds_load lookups, b128 vmem, and
// nontemporal stores for the streaming output.

typedef int    int4v   __attribute__((ext_vector_type(4)));
typedef float  float4v __attribute__((ext_vector_type(4)));

__device__ __forceinline__ void stage_lut_async(float* lut, const float* code, int tid) {
    // Each of the 256 lanes pulls one fp32 codebook entry straight into LDS
    // (no VGPR bounce); tracked by ASYNCcnt.
    unsigned lds_off = (unsigned)(size_t)lut + (unsigned)tid * 4u;
    const float* src = code + tid;
    asm volatile("global_load_async_to_lds_b32 %0, %1, off"
                 :: "v"(lds_off), "v"(src)
                 : "memory");
    asm volatile("s_wait_asynccnt 0" ::: "memory");
}

__device__ __forceinline__ float4v dequant4(int4v c, const float* lut, float scale) {
    float4v r;
    r.x = lut[c.x & 255] * scale;
    r.y = lut[c.y & 255] * scale;
    r.z = lut[c.z & 255] * scale;
    r.w = lut[c.w & 255] * scale;
    return r;
}

// Specialized D = 4096: fully unrolled, immediate offsets, 4 b128 loads in
// flight before any consumption.
__global__ __launch_bounds__(256) void frozen_bnb_embedding_d4096(
    const int*   __restrict__ input_ids,
    const int*   __restrict__ weight_codes,
    const float* __restrict__ absmax,
    const float* __restrict__ code,
    float*       __restrict__ out)
{
    constexpr int D = 4096;
    __shared__ float lut[256];

    const int tid = threadIdx.x;
    const int t   = blockIdx.x;

    stage_lut_async(lut, code, tid);
    __syncthreads();

    const int    row   = input_ids[t];          // uniform -> s_load
    const float  scale = absmax[row];           // uniform -> s_load
    const int*   wrow  = weight_codes + (size_t)row * (size_t)D + (size_t)(tid * 4);
    float*       orow  = out + (size_t)t * (size_t)D + (size_t)(tid * 4);

    // Issue all four 16B row loads first (4 KB/sweep across the block, fully
    // coalesced: consecutive lanes -> consecutive 16B chunks). The compiler
    // can overlap them and use partial loadcnt waits.
    int4v c0 = *(const int4v*)(wrow + 0 * 1024);
    int4v c1 = *(const int4v*)(wrow + 1 * 1024);
    int4v c2 = *(const int4v*)(wrow + 2 * 1024);
    int4v c3 = *(const int4v*)(wrow + 3 * 1024);

    // 16 ds_load_b32 lookups + v_mul_f32, then stream out with NT stores so
    // the 128 MB write doesn't evict anything useful from L2.
    __builtin_nontemporal_store(dequant4(c0, lut, scale), (float4v*)(orow + 0 * 1024));
    __builtin_nontemporal_store(dequant4(c1, lut, scale), (float4v*)(orow + 1 * 1024));
    __builtin_nontemporal_store(dequant4(c2, lut, scale), (float4v*)(orow + 2 * 1024));
    __builtin_nontemporal_store(dequant4(c3, lut, scale), (float4v*)(orow + 3 * 1024));
}

// Generic fallback for any D that is a multiple of 4.
__global__ __launch_bounds__(256) void frozen_bnb_embedding_generic(
    const int*   __restrict__ input_ids,
    const int*   __restrict__ weight_codes,
    const float* __restrict__ absmax,
    const float* __restrict__ code,
    float*       __restrict__ out,
    int D, int blocksize)
{
    __shared__ float lut[256];

    const int tid = threadIdx.x;
    const int t   = blockIdx.x;

    stage_lut_async(lut, code, tid);
    __syncthreads();

    const int    row  = input_ids[t];
    const int*   wrow = weight_codes + (size_t)row * (size_t)D;
    float*       orow = out + (size_t)t * (size_t)D;

    for (int idx = tid * 4; idx < D; idx += 256 * 4) {
        // absmax block index of flat element row*D + idx.
        const size_t flat  = (size_t)row * (size_t)D + (size_t)idx;
        const float  scale = absmax[flat / (size_t)blocksize];
        int4v c = *(const int4v*)(wrow + idx);
        __builtin_nontemporal_store(dequant4(c, lut, scale), (float4v*)(orow + idx));
    }
}

extern "C" void kernel_launch(void* const* d_in, const int* in_sizes, int n_in,
                              void* d_out, int out_size, void* d_ws, size_t ws_size,
                              hipStream_t stream) {
    const int*   input_ids    = (const int*)d_in[0];   // [B*S] = 8192
    const int*   weight_codes = (const int*)d_in[1];   // [V*D]
    const float* absmax       = (const float*)d_in[2]; // [V*D/BLOCKSIZE]
    const float* code         = (const float*)d_in[3]; // [256]
    float*       out          = (float*)d_out;

    const int n_tokens  = in_sizes[0];                 // 8192
    const int n_blocks  = in_sizes[2];                 // V*D/BLOCKSIZE
    const int D         = out_size / n_tokens;         // 4096
    const long long VD  = (long long)in_sizes[1];
    const int blocksize = (int)(VD / (long long)n_blocks); // 4096

    if (D == 4096 && blocksize == 4096) {
        frozen_bnb_embedding_d4096<<<n_tokens, 256, 0, stream>>>(
            input_ids, weight_codes, absmax, code, out);
    } else {
        frozen_bnb_embedding_generic<<<n_tokens, 256, 0, stream>>>(
            input_ids, weight_codes, absmax, code, out, D, blocksize);
    }
}